// GraphConvolutionSparse_36129264894615
// MI455X (gfx1250) — compile-verified
//
#include <hip/hip_runtime.h>

typedef float v2f __attribute__((ext_vector_type(2)));
typedef float v8f __attribute__((ext_vector_type(8)));

#define D_OUT 128

__global__ void gcn_zero4_kernel(float4* __restrict__ p, int n4) {
    int i = blockIdx.x * blockDim.x + threadIdx.x;
    int stride = gridDim.x * blockDim.x;
    for (; i < n4; i += stride) p[i] = make_float4(0.f, 0.f, 0.f, 0.f);
}

// COO scatter-SpMM with feature dim 128:
//   for each nonzero e: dst[rows[e], :] += vals[e] * src[cols[e], :]
// One wave32 processes a group of 16 nonzeros.
// Per 16-feature chunk j (8 chunks cover 128 features), accumulate 4
// V_WMMA_F32_16X16X4_F32 ops:
//   iteration t: A[M][k] = v_M iff (M == 4t+k), B[k][n] = src[c_{4t+k}][16j+n]
//   => D[M][n] = v_M * src[c_M][16j+n]   for M = 0..15
// then scatter D rows with f32 atomics (the true bottleneck; WMMA is free).
//
// Tail handling is data-masked, not exec-masked: invalid slots get v=0 so
// their D rows are exactly 0.0, and their scatter target is row 0 (valid
// memory). atomicAdd of 0.0f is inert modulo -0/+0, which the final ReLU
// erases — so every atomic is unconditional (no exec save/restore churn).
__global__ void gcn_coo_spmm_wmma(const float* __restrict__ vals,
                                  const int*   __restrict__ rows,
                                  const int*   __restrict__ cols,
                                  const float* __restrict__ src,
                                  float*       __restrict__ dst,
                                  int nnz) {
    const int lane  = threadIdx.x & 31;
    const int wave  = blockIdx.x * (blockDim.x >> 5) + (threadIdx.x >> 5);
    const int group = wave * 16;                 // wave-uniform
    if (group >= nnz) return;                    // uniform exit: EXEC stays all-1s

    const int m    = lane & 15;                  // row-in-group / column-in-tile
    const int half = lane >> 4;
    const int kx   = half ? 2 : 0;               // A/B K indices held by this lane
    const int ky   = kx + 1;

    const int  idx   = group + m;
    const bool valid = idx < nnz;
    const float v_mine = valid ? vals[idx] : 0.0f;
    const int   c_mine = valid ? cols[idx] : 0;
    const int   r_mine = valid ? rows[idx] : 0;

    // A operand for each of the 4 K-sweeps: masked diagonal of values.
    // ISA layout (32-bit A 16x4): lane<16 holds K=0,1; lane>=16 holds K=2,3; M=lane&15.
    v2f a[4];
#pragma unroll
    for (int t = 0; t < 4; ++t) {
        a[t].x = (((m >> 2) == t) && ((m & 3) == kx)) ? v_mine : 0.0f;
        a[t].y = (((m >> 2) == t) && ((m & 3) == ky)) ? v_mine : 0.0f;
    }

    // Source-row indices feeding B rows, broadcast across the wave.
    int cK0[4], cK1[4];
#pragma unroll
    for (int t = 0; t < 4; ++t) {
        cK0[t] = __shfl(c_mine, 4 * t + kx, 32);
        cK1[t] = __shfl(c_mine, 4 * t + ky, 32);
    }
    // Destination element offsets for the D tile rows this lane scatters.
    int dstOff[8];
#pragma unroll
    for (int r = 0; r < 8; ++r)
        dstOff[r] = __shfl(r_mine, r + 8 * half, 32) * D_OUT + m;

#pragma unroll
    for (int j = 0; j < 8; ++j) {                // 8 chunks of 16 features
        v8f acc = {};
#pragma unroll
        for (int t = 0; t < 4; ++t) {
            v2f b;
            // B layout (4x16 f32): lane<16 holds K=0,1 rows; lane>=16 holds K=2,3; N=lane&15.
            b.x = src[cK0[t] * D_OUT + j * 16 + m];
            b.y = src[cK1[t] * D_OUT + j * 16 + m];
            acc = __builtin_amdgcn_wmma_f32_16x16x4_f32(
                /*neg_a=*/false, a[t], /*neg_b=*/false, b,
                /*c_mod=*/(short)0, acc, /*reuse_a=*/false, /*reuse_b=*/false);
        }
        // D layout: VGPR r holds D[r + 8*half][n], n = lane&15.
#pragma unroll
        for (int r = 0; r < 8; ++r) {
            atomicAdd(&dst[dstOff[r] + j * 16], acc[r]);
        }
    }
}

__global__ void gcn_relu4_kernel(float4* __restrict__ p, int n4) {
    int i = blockIdx.x * blockDim.x + threadIdx.x;
    int stride = gridDim.x * blockDim.x;
    for (; i < n4; i += stride) {
        float4 v = p[i];
        v.x = fmaxf(v.x, 0.f); v.y = fmaxf(v.y, 0.f);
        v.z = fmaxf(v.z, 0.f); v.w = fmaxf(v.w, 0.f);
        p[i] = v;
    }
}

extern "C" void kernel_launch(void* const* d_in, const int* in_sizes, int n_in,
                              void* d_out, int out_size, void* d_ws, size_t ws_size,
                              hipStream_t stream) {
    // Inputs per reference order:
    const float* x_vals   = (const float*)d_in[0];
    const int*   x_rows   = (const int*)  d_in[1];
    const int*   x_cols   = (const int*)  d_in[2];
    const float* adj_vals = (const float*)d_in[3];
    const int*   adj_rows = (const int*)  d_in[4];
    const int*   adj_cols = (const int*)  d_in[5];
    const float* W        = (const float*)d_in[6];
    // d_in[7] = n_nodes (device scalar); host-side N derived from out_size.

    const int NNZ_X = in_sizes[0];
    const int N_E   = in_sizes[3];
    const int N     = out_size / D_OUT;
    const int hElems = N * D_OUT;                 // intermediate h = X @ W

    float* out = (float*)d_out;
    float* h   = (float*)d_ws;                    // needs N*128*4 = 51.2 MB scratch

    // Zero accumulators (harness poisons d_out/d_ws; atomics need clean zeros).
    {
        int n4 = hElems / 4;
        gcn_zero4_kernel<<<(n4 + 255) / 256, 256, 0, stream>>>((float4*)h, n4);
        int o4 = out_size / 4;
        gcn_zero4_kernel<<<(o4 + 255) / 256, 256, 0, stream>>>((float4*)out, o4);
    }

    // h = X @ W   (gather rows of W by feature column, scale, scatter-add by node row)
    {
        int waves  = (NNZ_X + 15) / 16;
        int blocks = (waves + 7) / 8;             // 256 threads = 8 waves/block
        gcn_coo_spmm_wmma<<<blocks, 256, 0, stream>>>(x_vals, x_rows, x_cols, W, h, NNZ_X);
    }

    // out = A @ h (h is L2-resident: 51.2 MB << 192 MB global L2)
    {
        int waves  = (N_E + 15) / 16;
        int blocks = (waves + 7) / 8;
        gcn_coo_spmm_wmma<<<blocks, 256, 0, stream>>>(adj_vals, adj_rows, adj_cols, h, out, N_E);
    }

    // out = relu(out)
    {
        int o4 = out_size / 4;
        gcn_relu4_kernel<<<(o4 + 255) / 256, 256, 0, stream>>>((float4*)out, o4);
    }
}